// HexDynamicConv_40647570489895
// MI455X (gfx1250) — compile-verified
//
#include <hip/hip_runtime.h>

typedef __attribute__((ext_vector_type(2))) float v2f;
typedef __attribute__((ext_vector_type(8))) float v8f;

#define BATCH 32
#define INCH  32
#define OUTCH 32
#define DYND  16
#define KNBR  7

// ---------------------------------------------------------------------------
// Kernel 1: build effective per-(b,k) weights, stored TRANSPOSED as
//   Wt[b][k][o][i] = Wstatic[k][i][o] + sum_d dyn[b][k][d] * Wdyn[k][d][i][o]
// One thread per (b,k,o,i) = 229376 threads. Negligible cost.
// ---------------------------------------------------------------------------
__global__ __launch_bounds__(256) void hexconv_build_weff(
    const float* __restrict__ dyn,      // (B,7,16)
    const float* __restrict__ Wdyn,     // (7,16,32,32)
    const float* __restrict__ Wstatic,  // (7,32,32)
    float* __restrict__ Wt)             // (B,7,32,32)  [o][i] order
{
    int t = blockIdx.x * blockDim.x + threadIdx.x;
    if (t >= BATCH * KNBR * OUTCH * INCH) return;
    int i = t & 31;
    int o = (t >> 5) & 31;
    int k = (t >> 10) % KNBR;
    int b = t / (1024 * KNBR);

    float s = Wstatic[(k * INCH + i) * OUTCH + o];
    const float* dv = dyn + (b * KNBR + k) * DYND;
    const float* wv = Wdyn + (((size_t)k * DYND) * INCH + i) * OUTCH + o; // stride over d = 1024
#pragma unroll
    for (int d = 0; d < DYND; ++d)
        s += dv[d] * wv[(size_t)d * INCH * OUTCH];

    Wt[((size_t)(b * KNBR + k) * OUTCH + o) * INCH + i] = s;
}

// ---------------------------------------------------------------------------
// Kernel 2: main hex conv via f32 WMMA (V_WMMA_F32_16X16X4_F32).
// blockIdx.y = batch, blockIdx.x covers n-tiles (8 waves/block, 1 tile/wave).
// A = gathered grid tile (16 rows x 32 in), B = Wt chunk (4 x 16 out).
// Per tile: 7 neighbors x 8 K-chunks x 2 out-halves = 112 WMMAs.
// ---------------------------------------------------------------------------
__global__ __launch_bounds__(256) void hexconv_wmma(
    const float* __restrict__ grid,   // (B,N,32)
    const int*   __restrict__ nbr,    // (7,N)
    const float* __restrict__ weff,   // (B,7,32,32) transposed [o][i]
    const float* __restrict__ bias,   // (32,)
    float*       __restrict__ out,    // (B,N,32)
    int N, int tilesPerB)
{
    const int lane = threadIdx.x & 31;
    const int wid  = threadIdx.x >> 5;
    const int nt   = blockIdx.x * 8 + wid;     // n-tile index (uniform per wave)
    if (nt >= tilesPerB) return;
    const int b  = blockIdx.y;
    const int n0 = nt * 16;

    const int col  = lane & 15;                // M-row / N-col index within tile
    const int half = lane >> 4;
    const int koff = half << 1;                // 0 or 2: K-offset of this lane's pair

    const int  row      = n0 + col;            // grid row this lane gathers (A matrix)
    const bool rowValid = row < N;

    const float* gb = grid + (size_t)b * N * INCH;
    const float* wb = weff + (size_t)b * KNBR * OUTCH * INCH;

    v8f c0 = {};                               // out channels 0..15
    v8f c1 = {};                               // out channels 16..31

    for (int k = 0; k < KNBR; ++k) {
        int idx = rowValid ? nbr[(size_t)k * N + row] : -1;
        int safe = idx >= 0 ? idx : 0;
        float m  = idx >= 0 ? 1.0f : 0.0f;     // branch-free mask: EXEC stays all-1s

        const float* gp = gb + (size_t)safe * INCH + koff;
        v2f a[8];
#pragma unroll
        for (int c = 0; c < 8; ++c) {
            v2f g = *(const v2f*)(gp + c * 4);  // contiguous 8B load from grid row
            a[c] = g * m;
        }

        const float* w0 = wb + (size_t)k * OUTCH * INCH + col * INCH + koff;  // o = col
        const float* w1 = w0 + 16 * INCH;                                     // o = col+16
#pragma unroll
        for (int c = 0; c < 8; ++c) {
            v2f b0 = *(const v2f*)(w0 + c * 4);
            v2f b1 = *(const v2f*)(w1 + c * 4);
            c0 = __builtin_amdgcn_wmma_f32_16x16x4_f32(false, a[c], false, b0,
                                                       (short)0, c0, false, false);
            c1 = __builtin_amdgcn_wmma_f32_16x16x4_f32(false, a[c], false, b1,
                                                       (short)0, c1, false, false);
        }
    }

    const float bl = bias[col];
    const float bh = bias[col + 16];
#pragma unroll
    for (int v = 0; v < 8; ++v) {
        int r = n0 + v + half * 8;             // C/D layout: VGPR v -> M = v (+8 upper lanes)
        if (r < N) {
            float* op = out + ((size_t)b * N + r) * OUTCH;
            op[col]      = c0[v] + bl;
            op[col + 16] = c1[v] + bh;
        }
    }
}

// ---------------------------------------------------------------------------
extern "C" void kernel_launch(void* const* d_in, const int* in_sizes, int n_in,
                              void* d_out, int out_size, void* d_ws, size_t ws_size,
                              hipStream_t stream) {
    const float* grid    = (const float*)d_in[0];
    const float* dyn     = (const float*)d_in[1];
    const int*   nbr     = (const int*)  d_in[2];
    const float* Wdyn    = (const float*)d_in[3];
    const float* Wstatic = (const float*)d_in[4];
    const float* bias    = (const float*)d_in[5];
    float*       out     = (float*)d_out;
    float*       weff    = (float*)d_ws;       // B*7*32*32 floats = 917504 B

    const int N = in_sizes[0] / (BATCH * INCH);

    const int total1 = BATCH * KNBR * OUTCH * INCH;
    hexconv_build_weff<<<(total1 + 255) / 256, 256, 0, stream>>>(dyn, Wdyn, Wstatic, weff);

    const int tilesPerB = (N + 15) / 16;
    dim3 gridDim((tilesPerB + 7) / 8, BATCH);
    hexconv_wmma<<<gridDim, 256, 0, stream>>>(grid, nbr, weff, bias, out, N, tilesPerB);
}